// VectorQuantizer_36429912605192
// MI455X (gfx1250) — compile-verified
//
#include <hip/hip_runtime.h>

// Problem constants (match the JAX reference)
static constexpr int EMB_D = 64;      // embedding dim
static constexpr int EMB_K = 8192;    // num codebook entries
static constexpr int EMB_N = 16384;   // 16*32*32 query vectors
static constexpr int KSPLIT = 4;      // codebook split for occupancy

typedef float v2f __attribute__((ext_vector_type(2)));
typedef float v4f __attribute__((ext_vector_type(4)));
typedef float v8f __attribute__((ext_vector_type(8)));
typedef int   v4i __attribute__((ext_vector_type(4)));

#if defined(__has_builtin)
#if __has_builtin(__builtin_amdgcn_global_load_async_to_lds_b128)
#define USE_ASYNC_LDS 1
#endif
#endif
#ifndef USE_ASYNC_LDS
#define USE_ASYNC_LDS 0
#endif

// ---------------------------------------------------------------------------
// Kernel 0: zero the one-hot output (512 MB) with non-temporal 128-bit stores
// ---------------------------------------------------------------------------
__global__ void k_zero(v4f* __restrict__ p, long n4) {
    long i = (long)blockIdx.x * blockDim.x + threadIdx.x;
    long stride = (long)gridDim.x * blockDim.x;
    v4f z = {0.f, 0.f, 0.f, 0.f};
    for (; i < n4; i += stride) {
        __builtin_nontemporal_store(z, p + i);
    }
}

// ---------------------------------------------------------------------------
// Kernel 1: cnorm[k] = ||codebook[k]||^2
// ---------------------------------------------------------------------------
__global__ void k_cnorm(const float* __restrict__ cb, float* __restrict__ cn) {
    int k = blockIdx.x * blockDim.x + threadIdx.x;
    if (k >= EMB_K) return;
    const v4f* row = (const v4f*)(cb + (size_t)k * EMB_D);
    float s = 0.f;
#pragma unroll
    for (int j = 0; j < EMB_D / 4; ++j) {
        v4f v = row[j];
        s += v[0]*v[0] + v[1]*v[1] + v[2]*v[2] + v[3]*v[3];
    }
    cn[k] = s;
}

// ---------------------------------------------------------------------------
// Async staging helpers: one 16B chunk per thread per tile (4KB tile / 256 thr)
// ---------------------------------------------------------------------------
__device__ __forceinline__ void stage_chunk(const float* gsrc, float* ldst) {
#if USE_ASYNC_LDS
    __builtin_amdgcn_global_load_async_to_lds_b128((v4i*)gsrc, (v4i*)ldst, 0, 0);
#else
    *(v4f*)ldst = *(const v4f*)gsrc;
#endif
}

__device__ __forceinline__ void stage_wait() {
#if USE_ASYNC_LDS
    asm volatile("s_wait_asynccnt 0" ::: "memory");
#endif
}

// ---------------------------------------------------------------------------
// Kernel 2: fused distance GEMM + running argmin via V_WMMA_F32_16X16X4_F32.
//   dist[m][n] = cnorm[n] - 2 * dot(x[m], c[n])      (||x||^2 dropped: row const)
// A fragments hold -2*x (folded scale), C initialized to cnorm broadcast, so
// the accumulator after the K=64 chain *is* the distance tile.
// Codebook tiles are staged cooperatively into LDS (double-buffered, async,
// 16B-padded rows -> bank-conflict-free ds_load_b64 B fragments).
// Each wave: 32 rows (two 16-row M fragments -> two independent WMMA chains).
// blockIdx.y selects one of KSPLIT codebook slices; partial (val,idx) -> ws.
// ---------------------------------------------------------------------------
__global__ void __launch_bounds__(256, 1)
k_argmin(const float* __restrict__ x, const float* __restrict__ cb,
         const float* __restrict__ cnorm,
         float* __restrict__ out_val, int* __restrict__ out_idx) {
    constexpr int ROW_F = 68;                       // 64 + 4 floats pad (bank shift 4)
    __shared__ alignas(16) float lds_tile[2][16 * ROW_F];

    const int tid  = threadIdx.x;
    const int lane = tid & 31;
    const int wave = tid >> 5;
    const int half = lane >> 4;       // selects even/odd K-pair per operand layout
    const int mrow = lane & 15;       // M (for A) / N (for B) index within tile
    const int row_base = blockIdx.x * 256 + wave * 32;
    const int ks = blockIdx.y;
    const int tiles = (EMB_K / 16) / KSPLIT;        // 128
    const int kt0 = ks * tiles;

    // staging coordinates: thread copies row srow, 16B chunk scol
    const int srow = tid >> 4;
    const int scol = (tid & 15) * 4;
    const float* gsrc0 = cb + ((size_t)(kt0 * 16) + srow) * EMB_D + scol;
    float* ldst[2] = { &lds_tile[0][srow * ROW_F + scol],
                       &lds_tile[1][srow * ROW_F + scol] };

    // ---- load A fragments (16x4 f32 each), scaled by -2, kept in VGPRs ----
    v2f a0[16], a1[16];
    const float* xr0 = x + (size_t)(row_base + mrow) * EMB_D + half * 2;
    const float* xr1 = x + (size_t)(row_base + 16 + mrow) * EMB_D + half * 2;
#pragma unroll
    for (int j = 0; j < 16; ++j) {
        v2f t0 = *(const v2f*)(xr0 + j * 4);
        v2f t1 = *(const v2f*)(xr1 + j * 4);
        a0[j] = -2.0f * t0;
        a1[j] = -2.0f * t1;
    }

    float minv[16];
    int   mini[16];
#pragma unroll
    for (int v = 0; v < 16; ++v) { minv[v] = __builtin_inff(); mini[v] = 0; }

    // prologue: stage tile 0
    stage_chunk(gsrc0, ldst[0]);
    stage_wait();
    __syncthreads();

    for (int t = 0; t < tiles; ++t) {
        const int cur = t & 1;
        // issue async stage of next tile into the other buffer
        if (t + 1 < tiles) {
            stage_chunk(gsrc0 + (size_t)(t + 1) * 16 * EMB_D, ldst[1 - cur]);
        }

        const int n_g = (kt0 + t) * 16 + mrow;      // this lane's code index
        const float cn = cnorm[n_g];

        const float* lbase = &lds_tile[cur][mrow * ROW_F + half * 2];
        v2f b[16];
#pragma unroll
        for (int j = 0; j < 16; ++j) b[j] = *(const v2f*)(lbase + j * 4);

        v8f acc0 = {cn, cn, cn, cn, cn, cn, cn, cn};
        v8f acc1 = acc0;
#pragma unroll
        for (int j = 0; j < 16; ++j) {
            acc0 = __builtin_amdgcn_wmma_f32_16x16x4_f32(
                false, a0[j], false, b[j], (short)0, acc0, false, false);
            acc1 = __builtin_amdgcn_wmma_f32_16x16x4_f32(
                false, a1[j], false, b[j], (short)0, acc1, false, false);
        }

        // running argmin: every acc component in this lane has N = n_g;
        // component v is row M = v + 8*half of the M-tile.
#pragma unroll
        for (int v = 0; v < 8; ++v) {
            float d0 = acc0[v];
            float d1 = acc1[v];
            if (d0 < minv[v])     { minv[v]     = d0; mini[v]     = n_g; }
            if (d1 < minv[8 + v]) { minv[8 + v] = d1; mini[8 + v] = n_g; }
        }

        if (t + 1 < tiles) {
            stage_wait();           // next tile's async copies landed in LDS
            __syncthreads();        // all waves done reading cur, staging done
        }
    }

    // ---- reduce (val,idx) across the 16 lanes of each half, first-min wins ----
#pragma unroll
    for (int v = 0; v < 16; ++v) {
        float mv = minv[v];
        int   mi = mini[v];
#pragma unroll
        for (int s = 1; s < 16; s <<= 1) {
            float ov = __shfl_xor(mv, s, 32);
            int   oi = __shfl_xor(mi, s, 32);
            if (ov < mv || (ov == mv && oi < mi)) { mv = ov; mi = oi; }
        }
        if (mrow == 0) {
            const int m = row_base + (v >> 3) * 16 + half * 8 + (v & 7);
            out_val[(size_t)ks * EMB_N + m] = mv;
            out_idx[(size_t)ks * EMB_N + m] = mi;
        }
    }
}

// ---------------------------------------------------------------------------
// Kernel 3: combine KSPLIT candidates, scatter the 1.0, copy quantized rows.
// 16 lanes per row; 128-bit copies for the codebook row.
// ---------------------------------------------------------------------------
__global__ void k_scatter(const float* __restrict__ val, const int* __restrict__ idx,
                          const float* __restrict__ cb,
                          float* __restrict__ disc, float* __restrict__ quant) {
    const int g = threadIdx.x >> 4;           // 16 row-groups per block
    const int l = threadIdx.x & 15;
    const int r = blockIdx.x * 16 + g;
    if (r >= EMB_N) return;

    float bv = val[r];
    int   bi = idx[r];
#pragma unroll
    for (int s = 1; s < KSPLIT; ++s) {
        float v = val[(size_t)s * EMB_N + r];
        int   i = idx[(size_t)s * EMB_N + r];
        if (v < bv) { bv = v; bi = i; }       // ties keep lower split == lower idx
    }
    if (l == 0) disc[(size_t)r * EMB_K + bi] = 1.0f;
    v4f c = *(const v4f*)(cb + (size_t)bi * EMB_D + l * 4);
    *(v4f*)(quant + (size_t)r * EMB_D + l * 4) = c;
}

// ---------------------------------------------------------------------------
extern "C" void kernel_launch(void* const* d_in, const int* in_sizes, int n_in,
                              void* d_out, int out_size, void* d_ws, size_t ws_size,
                              hipStream_t stream) {
    (void)in_sizes; (void)n_in; (void)out_size; (void)ws_size;

    const float* x  = (const float*)d_in[0];   // [N, 64]
    const float* cb = (const float*)d_in[1];   // [K, 64]

    float* disc  = (float*)d_out;                          // [N, K]
    float* quant = disc + (size_t)EMB_N * EMB_K;           // [N, 64]

    // workspace layout: per-split (val, idx) candidates + cnorm
    float* ws_val  = (float*)d_ws;                                   // KSPLIT*N f32
    int*   ws_idx  = (int*)(ws_val + (size_t)KSPLIT * EMB_N);        // KSPLIT*N i32
    float* ws_cn   = (float*)(ws_idx + (size_t)KSPLIT * EMB_N);      // K f32

    // 1) zero the one-hot output (streaming stores)
    long n4 = (long)EMB_N * EMB_K / 4;
    k_zero<<<8192, 256, 0, stream>>>((v4f*)disc, n4);

    // 2) codebook squared norms
    k_cnorm<<<EMB_K / 256, 256, 0, stream>>>(cb, ws_cn);

    // 3) fused distance GEMM + argmin (WMMA f32 16x16x4, async LDS staging)
    dim3 grid(EMB_N / 256, KSPLIT);
    k_argmin<<<grid, 256, 0, stream>>>(x, cb, ws_cn, ws_val, ws_idx);

    // 4) combine splits, scatter ones, gather quantized rows
    k_scatter<<<EMB_N / 16, 256, 0, stream>>>(ws_val, ws_idx, cb, disc, quant);
}